// LI_55035710931679
// MI455X (gfx1250) — compile-verified
//
#include <hip/hip_runtime.h>

#define L_SEQ  16384
#define H_DIM  1024
#define CHUNK  256                 // 16 sub-chunks of 16 steps, handled by WMMA
#define NCHUNK (L_SEQ / CHUNK)     // 64
#define LDS_PITCH 33               // 32 channels + 1 pad word

typedef float v2f __attribute__((ext_vector_type(2)));
typedef float v8f __attribute__((ext_vector_type(8)));

// ---------------------------------------------------------------------------
// Kernel 1: per-chunk local aggregates  E[c][h] = zero-init scan end of chunk c
// Coalesced: 256 consecutive channels per block, marching over time.
// ---------------------------------------------------------------------------
__global__ void ssm_chunk_agg(const float* __restrict__ x,
                              const float* __restrict__ tau,
                              float* __restrict__ E) {
    const int h  = blockIdx.x * blockDim.x + threadIdx.x;   // 0..H-1
    const int c  = blockIdx.y;                              // 0..NCHUNK-1
    const int t0 = c * CHUNK;
    const float a = expf(tau[h]);
    float e = 0.0f;
#pragma unroll 8
    for (int i = 0; i < CHUNK; ++i)
        e = fmaf(a, e, x[(t0 + i) * H_DIM + h]);
    E[c * H_DIM + h] = e;
}

// ---------------------------------------------------------------------------
// Kernel 2: exclusive scan of chunk aggregates per channel.
// Cin[c][h] = inclusive state at time (c*CHUNK - 1); Cin[0] = 0.
// ---------------------------------------------------------------------------
__global__ void ssm_chunk_carry(const float* __restrict__ E,
                                const float* __restrict__ tau,
                                float* __restrict__ Cin) {
    const int h = blockIdx.x * blockDim.x + threadIdx.x;    // 0..H-1
    const float bC = expf(tau[h] * (float)CHUNK);           // a^CHUNK
    float s = 0.0f;
    for (int c = 0; c < NCHUNK; ++c) {
        Cin[c * H_DIM + h] = s;
        s = fmaf(bC, s, E[c * H_DIM + h]);
    }
}

// ---------------------------------------------------------------------------
// Kernel 3: apply. One 1024-thread block = 32 waves = 32 channels x 256 steps.
// Per wave (one channel): Z(16x16) = Lmat(16x16) @ X(16x16) via 4x WMMA f32
// 16x16x4, where X[s][j] = x[t0 + 16j + s] and Lmat[t][s] = a^(t-s) (lower
// triangular). Then inter-sub-chunk carries
//   d[j] = b^j*carry + sum_{k<j} b^(j-1-k) * Z[15][k]   (b = a^16)
// via readlane, and y[16j+t] = Z[t][j] + a^(t+1) * d[j]. LDS-staged so all
// global traffic is 128B-coalesced across 32 consecutive channels.
// All decay powers are generated multiplicatively from a single expf per
// operand family -> branchless v_cndmask masking, EXEC stays all-ones.
// ---------------------------------------------------------------------------
__global__ void __launch_bounds__(1024)
ssm_apply_wmma(const float* __restrict__ x,
               const float* __restrict__ tau,
               const float* __restrict__ Cin,
               float* __restrict__ out) {
    __shared__ float tile[CHUNK * LDS_PITCH];

    const int tid  = threadIdx.x;
    const int wv   = tid >> 5;          // wave id == channel-in-group
    const int lane = tid & 31;
    const int j    = lane & 15;         // N index (sub-chunk) / M index for A
    const int half = lane >> 4;
    const int ch2  = tid & 31;          // channel for cooperative ld/st
    const int h0   = blockIdx.x * 32;
    const int c    = blockIdx.y;
    const int t0   = c * CHUNK;

    // ---- stage 1: cooperative coalesced load of (256 time x 32 ch) tile ----
#pragma unroll
    for (int r = 0; r < 8; ++r) {
        const int t = r * 32 + wv;
        tile[t * LDS_PITCH + ch2] = x[(t0 + t) * H_DIM + h0 + ch2];
    }
    __syncthreads();

    // ---- stage 2: per-wave WMMA chunked scan for channel h0+wv ----
    const int   h     = h0 + wv;
    const float tau_h = tau[h];
    const float a     = expf(tau_h);
    const float ainv  = 1.0f / a;
    const float ainv2 = ainv * ainv;
    const float ainv4 = ainv2 * ainv2;
    const float carry = Cin[c * H_DIM + h];   // wave-uniform

    // Gather B operands (X columns for this channel) from LDS first.
    float bxv[4], byv[4];
#pragma unroll
    for (int kk = 0; kk < 4; ++kk) {
        const int k0 = 4 * kk + 2 * half;           // K index of B vgpr0
        bxv[kk] = tile[(16 * j + k0    ) * LDS_PITCH + wv];
        byv[kk] = tile[(16 * j + k0 + 1) * LDS_PITCH + wv];
    }

    // A-matrix powers: at slice kk, av.x = a^(d0), av.y = a^(d0-1) masked,
    // d0 = j - (4kk + 2*half). Seed with one per-lane expf, step by a^-4.
    float px = expf(tau_h * (float)(j - 2 * half));  // a^(j-2*half)
    float py = px * ainv;

    v8f acc = {0.f, 0.f, 0.f, 0.f, 0.f, 0.f, 0.f, 0.f};
#pragma unroll
    for (int kk = 0; kk < 4; ++kk) {
        const int d0 = j - (4 * kk + 2 * half);      // m - k for A vgpr0
        v2f av, bv;
        av.x = (d0 >= 0) ? px : 0.0f;                // branchless cndmask
        av.y = (d0 >= 1) ? py : 0.0f;
        bv.x = bxv[kk];
        bv.y = byv[kk];
        // D = A(16x4) * B(4x16) + C, full f32
        acc = __builtin_amdgcn_wmma_f32_16x16x4_f32(
                  false, av, false, bv, (short)0, acc, false, false);
        px *= ainv4;
        py *= ainv4;
    }

    // ---- cross-sub-chunk carry chain: d[j] for this lane's column ----
    // zend[k] = Z[15][k] lives in acc[7] on lanes 16+k.
    const float binv16 = expf(-16.0f * tau_h);        // b^-1, b = a^16
    const float powb_j = expf(16.0f * tau_h * (float)j); // b^j
    float q = powb_j * binv16;                        // b^(j-1)
    float d = powb_j * carry;                         // b^j * carry_in
#pragma unroll
    for (int k = 0; k < 15; ++k) {
        const float zk = __int_as_float(
            __builtin_amdgcn_readlane(__float_as_int(acc[7]), 16 + k));
        d = fmaf((j > k) ? q : 0.0f, zk, d);
        q *= binv16;
    }

    __syncthreads();   // all waves done reading the input tile

    // ---- write corrected Y back into the tile (time-major, ch minor) ----
    // pa = a^(m+1) with m = v + 8*half
    const float a2 = a * a;
    const float a4 = a2 * a2;
    const float a8 = a4 * a4;
    float pa = a * (half ? a8 : 1.0f);
#pragma unroll
    for (int v = 0; v < 8; ++v) {
        const int m = v + 8 * half;                  // row of D this vgpr holds
        const float y = fmaf(pa, d, acc[v]);
        tile[(16 * j + m) * LDS_PITCH + wv] = y;
        pa *= a;
    }
    __syncthreads();

    // ---- stage 3: cooperative coalesced store ----
#pragma unroll
    for (int r = 0; r < 8; ++r) {
        const int t = r * 32 + wv;
        out[(t0 + t) * H_DIM + h0 + ch2] = tile[t * LDS_PITCH + ch2];
    }
}

// ---------------------------------------------------------------------------
extern "C" void kernel_launch(void* const* d_in, const int* in_sizes, int n_in,
                              void* d_out, int out_size, void* d_ws, size_t ws_size,
                              hipStream_t stream) {
    (void)in_sizes; (void)n_in; (void)out_size; (void)ws_size;
    const float* x   = (const float*)d_in[0];   // (L, H) f32
    const float* tau = (const float*)d_in[1];   // (H,)  f32
    float*       y   = (float*)d_out;           // (L, H) f32

    float* E   = (float*)d_ws;                  // NCHUNK*H chunk aggregates
    float* Cin = E + NCHUNK * H_DIM;            // NCHUNK*H chunk carries

    // Phase 1: per-chunk aggregates (reads all of x -> warms 192MB L2)
    ssm_chunk_agg<<<dim3(H_DIM / 256, NCHUNK), dim3(256), 0, stream>>>(x, tau, E);
    // Phase 2: tiny cross-chunk scan
    ssm_chunk_carry<<<dim3(H_DIM / 256), dim3(256), 0, stream>>>(E, tau, Cin);
    // Phase 3: WMMA chunked-scan apply (x re-read hits L2; writes y)
    ssm_apply_wmma<<<dim3(H_DIM / 32, NCHUNK), dim3(1024), 0, stream>>>(x, tau, Cin, y);
}